// ActiveDiscriminator_46840913330615
// MI455X (gfx1250) — compile-verified
//
#include <hip/hip_runtime.h>
#include <hip/hip_bf16.h>
#include <math.h>

// ---------------------------------------------------------------------------
// ActiveDiscriminator: score-weighted segmented mean (N=262144, D=256, B=16)
// followed by 16x256 dot + sigmoid.
//
// MI455X strategy:
//   segment_sum == GEMM: sums(16x256) = [OneHot(seg)*score](16xN) x PF(Nx256)
//   via V_WMMA_F32_16X16X4_F32 (wave32). A = one-hot(seg)*score fuses the
//   score scaling AND the segment scatter into the matrix op. PF (256 MB) is
//   streamed once with NT loads -> ~11.5 us HBM roofline; no data reuse, so
//   direct-to-VGPR beats any LDS/TDM staging.
//
//   Counts exploit that point_coords[:,0] is SORTED: 17 binary searches give
//   all segment boundaries, removing all counting work from the hot loop.
//
//   Hot loop uses a compile-time trip count (128) so the compiler emits a
//   scalar-controlled, unrolled loop (no divergent exec-mask loop).
// ---------------------------------------------------------------------------

typedef __attribute__((ext_vector_type(2))) float v2f;
typedef __attribute__((ext_vector_type(8))) float v8f;

#define NPTS      262144
#define DFEAT     256
#define NSEG      16
#define NCHUNK    512                 // point chunks
#define PTS_PER_CHUNK (NPTS / NCHUNK) // 512
#define ITERS     (PTS_PER_CHUNK / 4) // 128 wmma steps per wave
#define NTILE     (DFEAT / 16)        // 16 feature tiles of 16 cols
#define WAVES_PER_BLOCK 8
#define TOTAL_WAVES (NTILE * NCHUNK)  // 8192
#define MAIN_BLOCKS (TOTAL_WAVES / WAVES_PER_BLOCK) // 1024

// ws layout: sums[16*256] floats, then counts[16] floats
#define WS_SUMS_ELEMS (NSEG * DFEAT)

__global__ void ad_init_sums(float* sums) {
    int i = blockIdx.x * blockDim.x + threadIdx.x;
    if (i < WS_SUMS_ELEMS) sums[i] = 0.0f;
}

// point_coords[:,0] is sorted ascending -> counts via segment boundaries.
__global__ void ad_counts(const int* __restrict__ coords,
                          float* __restrict__ counts) {
    __shared__ int lb[NSEG + 1];
    const int t = threadIdx.x;
    if (t <= NSEG) {
        int lo = 0, hi = NPTS;          // lower_bound of segment id t
        while (lo < hi) {
            const int mid = (lo + hi) >> 1;
            if (coords[mid * 4] < t) lo = mid + 1; else hi = mid;
        }
        lb[t] = lo;
    }
    __syncthreads();
    if (t < NSEG) counts[t] = (float)(lb[t + 1] - lb[t]);
}

__global__ __launch_bounds__(256) void ad_segsum_wmma(
    const float* __restrict__ pf,      // [N, 256]
    const float* __restrict__ scores,  // [N]
    const int*   __restrict__ coords,  // [N, 4], col 0 = segment id
    float*       __restrict__ sums)    // [16, 256]
{
    const int lane  = threadIdx.x & 31;
    const int wib   = threadIdx.x >> 5;
    const int gwave = blockIdx.x * WAVES_PER_BLOCK + wib;

    const int tile  = gwave & (NTILE - 1);   // feature cols [tile*16, tile*16+16)
    const int chunk = gwave >> 4;            // point chunk

    const int half = lane >> 4;              // 0: lanes 0-15, 1: lanes 16-31
    const int m    = lane & 15;              // A row (segment) / col within tile
    const int col  = tile * 16 + m;          // feature column owned by this lane

    // ISA A-layout 16x4 f32: lanes 0-15 hold K=0,1 (VGPR0,1); lanes 16-31
    // hold K=2,3. B (4x16) mirrors this. This lane's K pair starts at half*2.
    const int p0 = chunk * PTS_PER_CHUNK + half * 2;

    const float* pfp = pf + (size_t)p0 * DFEAT + col; // rows k0, k0+1 via +DFEAT
    const int*   cp  = coords + p0 * 4;
    const float* sp  = scores + p0;                   // even index -> 8B aligned

    v8f acc = {};  // C/D: 16x16 f32 accumulator (8 VGPRs)

    #pragma unroll 4
    for (int i = 0; i < ITERS; ++i) {     // compile-time trip count: 128
        v2f b;
        b.x = __builtin_nontemporal_load(pfp);
        b.y = __builtin_nontemporal_load(pfp + DFEAT);

        const int seg0 = cp[0];
        const int seg1 = cp[4];
        const v2f s01  = *(const v2f*)sp; // one b64 load for both scores

        v2f a;
        a.x = (seg0 == m) ? s01.x : 0.0f;
        a.y = (seg1 == m) ? s01.y : 0.0f;

        acc = __builtin_amdgcn_wmma_f32_16x16x4_f32(
            /*neg_a=*/false, a, /*neg_b=*/false, b,
            /*c_mod=*/(short)0, acc, /*reuse_a=*/false, /*reuse_b=*/false);

        pfp += 4 * DFEAT;
        cp  += 4 * 4;
        sp  += 4;
    }

    // Flush: C/D layout -> VGPR r holds row M = r + 8*half, col N = m.
    #pragma unroll
    for (int r = 0; r < 8; ++r) {
        const int row = r + half * 8;
        atomicAdd(&sums[row * DFEAT + col], acc[r]);
    }
}

__global__ __launch_bounds__(256) void ad_finalize(
    const float* __restrict__ sums,    // [16, 256]
    const float* __restrict__ counts,  // [16]
    const float* __restrict__ w,       // [256] (fc_weight, out_features=1)
    const float* __restrict__ bias,    // [1]
    float*       __restrict__ out)     // [16]
{
    __shared__ float red[256];
    const int t = threadIdx.x;
    const float wt = w[t];

    for (int b = 0; b < NSEG; ++b) {
        red[t] = (sums[b * DFEAT + t] / counts[b]) * wt;
        __syncthreads();
        for (int s = 128; s > 0; s >>= 1) {
            if (t < s) red[t] += red[t + s];
            __syncthreads();
        }
        if (t == 0) {
            const float x = red[0] + bias[0];
            out[b] = 1.0f / (1.0f + __expf(-x));
        }
        __syncthreads();
    }
}

extern "C" void kernel_launch(void* const* d_in, const int* in_sizes, int n_in,
                              void* d_out, int out_size, void* d_ws, size_t ws_size,
                              hipStream_t stream) {
    const float* pf     = (const float*)d_in[0]; // point_features [N,256]
    const float* scores = (const float*)d_in[1]; // point_cls_scores [N]
    const int*   coords = (const int*)  d_in[2]; // point_coords [N,4]
    const float* fw     = (const float*)d_in[3]; // fc_weight [1,256]
    const float* fb     = (const float*)d_in[4]; // fc_bias [1]
    float*       out    = (float*)d_out;         // [16,1]

    float* sums   = (float*)d_ws;
    float* counts = sums + WS_SUMS_ELEMS;

    ad_init_sums<<<(WS_SUMS_ELEMS + 255) / 256, 256, 0, stream>>>(sums);

    ad_counts<<<1, 32, 0, stream>>>(coords, counts);

    ad_segsum_wmma<<<MAIN_BLOCKS, 256, 0, stream>>>(pf, scores, coords, sums);

    ad_finalize<<<1, 256, 0, stream>>>(sums, counts, fw, fb, out);
}